// AdaptiveSam_55851754717601
// MI455X (gfx1250) — compile-verified
//
#include <hip/hip_runtime.h>
#include <hip/hip_bf16.h>

#define OUT_HW 224
#define IMG_H  1024
#define IMG_W  1024
#define CIN    3
#define COUT   3
#define NTILE  (OUT_HW / 16)     // 14 pixel tiles of 16 per output row
#define PROW   (IMG_W + 8)       // padded row stride: 4 zero cols each side

typedef __attribute__((ext_vector_type(2))) float v2f;
typedef __attribute__((ext_vector_type(8))) float v8f;

// Builtin signature (from hipcc diagnostic): vV4i*1 V4i*3 Ii Ii
typedef int v4i_g __attribute__((vector_size(16)));
typedef __attribute__((address_space(1))) v4i_g* as1_v4ip;
typedef __attribute__((address_space(3))) v4i_g* as3_v4ip;

#if defined(__has_builtin)
#  if __has_builtin(__builtin_amdgcn_global_load_async_to_lds_b128)
#    define HAVE_ASYNC_B128 1
#  endif
#  if __has_builtin(__builtin_amdgcn_s_wait_asynccnt)
#    define HAVE_WAIT_ASYNC 1
#  endif
#  if __has_builtin(__builtin_amdgcn_wmma_f32_16x16x4_f32)
#    define HAVE_WMMA_F32_K4 1
#  endif
#endif

__device__ __forceinline__ void copy16_g2lds(const float* __restrict__ g, float* l) {
#if defined(HAVE_ASYNC_B128)
  __builtin_amdgcn_global_load_async_to_lds_b128((as1_v4ip)g, (as3_v4ip)l, 0, 0);
#else
  float4 v = *(const float4*)g;
  *(float4*)l = v;
#endif
}

__device__ __forceinline__ void drain_async_then_barrier() {
#if defined(HAVE_ASYNC_B128)
#  if defined(HAVE_WAIT_ASYNC)
  __builtin_amdgcn_s_wait_asynccnt(0);
#  else
  asm volatile("s_wait_asynccnt 0" ::: "memory");
#  endif
#endif
  __syncthreads();
}

// Two-way select that the compiler cannot fold back into a dynamically indexed
// load from a private array: pin both candidates as register values first.
__device__ __forceinline__ float sel2(bool c, float a, float b) {
  asm("" : "+v"(a), "+v"(b));
  return c ? a : b;
}

__global__ __launch_bounds__(256, 1)
void adaptive_sam_kernel(const float* __restrict__ x,
                         const float* __restrict__ stride_h,
                         const float* __restrict__ stride_w,
                         const float* __restrict__ weight,
                         float* __restrict__ out)
{
  // 12 padded rows (3ch x 4 rows x 1032 cols) ~48.4 KB + padded weight 16x28.
  __shared__ float patch[12 * PROW];
  __shared__ float Wpad[16 * 28];

  const int tid = threadIdx.x;
  const int i   = blockIdx.x;   // output row
  const int b   = blockIdx.y;   // batch

  // ---- padded weight Wpad[m][k]: m = out channel (3->16), k = c*9+u*3+v (27->28)
  for (int t = tid; t < 16 * 28; t += 256) {
    const int m = t / 28, k = t % 28;
    Wpad[t] = (m < COUT && k < 27) ? weight[m * 27 + k] : 0.0f;
  }
  // ---- zero the 4-col pads on both sides of each of the 12 rows
  if (tid < 12 * 8) {
    const int r = tid >> 3, s = tid & 7;
    patch[r * PROW + ((s < 4) ? s : (1024 + s))] = 0.0f;   // cols 0..3 and 1028..1031
  }

  // ---- y-axis bilinear terms (shared by the whole output row)
  const float cy = (stride_h[b * OUT_HW + i] + 1.0f) * (0.5f * (IMG_H - 1));
  const float fy = cy - floorf(cy);
  const int   iy = (int)floorf(cy);

  // ---- async-stage 12 image rows into LDS (each thread: one float4 per row)
  {
    const int col4 = tid * 4;  // 256 threads * 4 floats == 1024 == IMG_W
#pragma unroll
    for (int r12 = 0; r12 < 12; ++r12) {
      const int c  = r12 >> 2;
      const int ty = r12 & 3;
      const int rr = iy - 1 + ty;
      const int rc = min(max(rr, 0), IMG_H - 1);   // clamped; invalid rows get zero weight
      const float* gsrc = x + (((size_t)b * CIN + c) * IMG_H + rc) * IMG_W + col4;
      copy16_g2lds(gsrc, &patch[r12 * PROW + 4 + col4]);
    }
  }
  drain_async_then_barrier();

  // ---- per-wave setup
  const int lane = tid & 31;
  const int wv   = tid >> 5;
  const int p16  = lane & 15;         // matrix column: pixel within tile / out-channel row
  const bool lo  = (lane < 16);       // lane half selects K pair {0,1} vs {2,3}
  const int ksel = lo ? 0 : 2;

  // A fragments (weight) are loop-invariant: hoist. A 16x4 layout per kb:
  // VGPR0 = W[m][kb+ksel], VGPR1 = W[m][kb+ksel+1], m = p16.
  float aX[7], aY[7];
#pragma unroll
  for (int t = 0; t < 7; ++t) {
    aX[t] = Wpad[p16 * 28 + 4 * t + ksel];
    aY[t] = Wpad[p16 * 28 + 4 * t + ksel + 1];
  }

  // row weights: wy0[u] pairs with patch row ty=u, wy1[u] with ty=u+1
  float wy0u[3], wy1u[3];
#pragma unroll
  for (int u = 0; u < 3; ++u) {
    const int r0 = iy + u - 1, r1 = iy + u;
    wy0u[u] = (1.0f - fy) * ((r0 >= 0 && r0 < IMG_H) ? 1.0f : 0.0f);
    wy1u[u] = fy          * ((r1 >= 0 && r1 < IMG_H) ? 1.0f : 0.0f);
  }

  for (int tile = wv; tile < NTILE; tile += 8) {   // wave-uniform loop: EXEC full at WMMA
    const int jp = tile * 16 + p16;

    // x-axis bilinear terms; padded rows make clamping/masking unnecessary
    const float cx = (stride_w[b * OUT_HW + jp] + 1.0f) * (0.5f * (IMG_W - 1));
    const float fx = cx - floorf(cx);
    const int   ix = (int)floorf(cx);
    const int   xbase = ix + 3;       // patch col of tap tx=0 (= ix-1 plus 4-col pad)
    const float wx0 = 1.0f - fx, wx1 = fx;

    // sampled values val[k], k = c*9 + u*3 + v  (27 real + 1 zero pad)
    float val[28];
    val[27] = 0.0f;
#pragma unroll
    for (int c = 0; c < CIN; ++c) {
      float P[4][4];
#pragma unroll
      for (int ty = 0; ty < 4; ++ty)
#pragma unroll
        for (int tx = 0; tx < 4; ++tx)
          P[ty][tx] = patch[(c * 4 + ty) * PROW + xbase + tx];

      float tmp[4][3];
#pragma unroll
      for (int ty = 0; ty < 4; ++ty)
#pragma unroll
        for (int v = 0; v < 3; ++v)
          tmp[ty][v] = fmaf(wx0, P[ty][v], wx1 * P[ty][v + 1]);

#pragma unroll
      for (int u = 0; u < 3; ++u)
#pragma unroll
        for (int v = 0; v < 3; ++v)
          val[c * 9 + u * 3 + v] = fmaf(wy0u[u], tmp[u][v], wy1u[u] * tmp[u + 1][v]);
    }

#if defined(HAVE_WMMA_F32_K4)
    // out[o, pixel] = sum_k W[o,k] * val[k, pixel] via 7x V_WMMA_F32_16X16X4_F32.
    // B 4x16 layout per kb: VGPR0 = val[kb+ksel], VGPR1 = val[kb+ksel+1], col = p16.
    v8f acc = {};
#pragma unroll
    for (int t = 0; t < 7; ++t) {
      const int kb = 4 * t;
      v2f a, bm;
      a.x  = aX[t];
      a.y  = aY[t];
      bm.x = sel2(lo, val[kb + 0], val[kb + 2]);
      bm.y = sel2(lo, val[kb + 1], val[kb + 3]);
      acc = __builtin_amdgcn_wmma_f32_16x16x4_f32(false, a, false, bm,
                                                  (short)0, acc, false, false);
    }
    if (lo) {  // D VGPR r, lanes 0-15 hold (M=r, N=lane): rows 0..2 = real out channels
      const size_t base = (((size_t)b * COUT) * OUT_HW + i) * OUT_HW + jp;
      out[base + 0 * (size_t)OUT_HW * OUT_HW] = acc[0];
      out[base + 1 * (size_t)OUT_HW * OUT_HW] = acc[1];
      out[base + 2 * (size_t)OUT_HW * OUT_HW] = acc[2];
    }
#else
    float o0 = 0.0f, o1 = 0.0f, o2 = 0.0f;
#pragma unroll
    for (int k = 0; k < 27; ++k) {
      const float v = val[k];
      o0 = fmaf(Wpad[0 * 28 + k], v, o0);
      o1 = fmaf(Wpad[1 * 28 + k], v, o1);
      o2 = fmaf(Wpad[2 * 28 + k], v, o2);
    }
    if (lo) {
      const size_t base = (((size_t)b * COUT) * OUT_HW + i) * OUT_HW + jp;
      out[base + 0 * (size_t)OUT_HW * OUT_HW] = o0;
      out[base + 1 * (size_t)OUT_HW * OUT_HW] = o1;
      out[base + 2 * (size_t)OUT_HW * OUT_HW] = o2;
    }
#endif
  }
}

extern "C" void kernel_launch(void* const* d_in, const int* in_sizes, int n_in,
                              void* d_out, int out_size, void* d_ws, size_t ws_size,
                              hipStream_t stream) {
  const float* x  = (const float*)d_in[0];
  const float* sh = (const float*)d_in[1];
  const float* sw = (const float*)d_in[2];
  const float* wg = (const float*)d_in[3];
  float* out = (float*)d_out;

  const int B = in_sizes[1] / OUT_HW;       // 16
  dim3 grid(OUT_HW, B);                     // one block per (i, b) output row
  adaptive_sam_kernel<<<grid, 256, 0, stream>>>(x, sh, sw, wg, out);
}